// ClassicalRBFFeature_65481071405915
// MI455X (gfx1250) — compile-verified
//
#include <hip/hip_runtime.h>

typedef float v2f __attribute__((ext_vector_type(2)));
typedef float v4f __attribute__((ext_vector_type(4)));
typedef float v8f __attribute__((ext_vector_type(8)));

#define N_ROWS 32768
#define M_COLS 2048
#define KDIM   64
#define BLK    64                  // 64x64 output per wave (4x4 WMMA tiles)
#define ROW_BLKS (N_ROWS / BLK)    // 512
#define COL_BLKS (M_COLS / BLK)    // 32
#define WAVES_PER_BLOCK 8

// ---- Pass 1: per-row sum of squares (for x and for reference_points) ----
__global__ __launch_bounds__(256) void rbf_sumsq_kernel(const float* __restrict__ in,
                                                        float* __restrict__ out,
                                                        int rows) {
    int i = blockIdx.x * blockDim.x + threadIdx.x;
    if (i >= rows) return;
    const float4* p = (const float4*)(in + (size_t)i * KDIM);
    float s = 0.f;
#pragma unroll
    for (int j = 0; j < KDIM / 4; ++j) {
        float4 v = p[j];
        s = fmaf(v.x, v.x, s);
        s = fmaf(v.y, v.y, s);
        s = fmaf(v.z, v.z, s);
        s = fmaf(v.w, v.w, s);
    }
    out[i] = s;
}

// ---- Pass 2: 4x4 register-blocked WMMA f32 GEMM + RBF epilogue ----
// One wave owns a 64x64 output block. Per k-step (K granularity 4):
//   4 A fragments (b64 loads, row tiles at +4096B const offsets)
//   4 B fragments (b128 load of K[k0..k0+3], select {half, half+2})
//   16 chained V_WMMA_F32_16X16X4_F32 accumulations.
__global__ __launch_bounds__(WAVES_PER_BLOCK * 32) void
rbf_wmma_kernel(const float* __restrict__ x,
                const float* __restrict__ r,
                const float* __restrict__ xsq,
                const float* __restrict__ rsq,
                float* __restrict__ out) {
    const int lane = threadIdx.x & 31;
    const int wave = threadIdx.x >> 5;

    // Col blocks innermost: the 8 waves of a workgroup share A rows in WGP$.
    const unsigned t = blockIdx.x * WAVES_PER_BLOCK + wave;
    const int colBlk = t & (COL_BLKS - 1);
    const int rowBlk = t >> 5;               // / COL_BLKS
    const int rowBase = rowBlk * BLK;
    const int colBase = colBlk * BLK;

    const int half = lane >> 4;              // 0: lanes 0-15, 1: lanes 16-31
    const int l    = lane & 15;

    // A: row = rowBase + i*16 + l, K pair = k0 + 2*half + {0,1} (contiguous b64)
    const float* aptr = x + (size_t)(rowBase + l) * KDIM + 2 * half;
    // B: col = colBase + j*16 + l; load K[k0..k0+3] as b128, lane uses {half, half+2}
    const float* bptr = r + (size_t)(colBase + l) * KDIM;

    v8f acc[4][4] = {};

#pragma unroll 2
    for (int k0 = 0; k0 < KDIM; k0 += 4) {
        v2f a[4];
#pragma unroll
        for (int i = 0; i < 4; ++i)
            a[i] = *(const v2f*)(aptr + i * 16 * KDIM + k0);

        v2f b[4];
#pragma unroll
        for (int j = 0; j < 4; ++j) {
            v4f q = *(const v4f*)(bptr + j * 16 * KDIM + k0);
            v2f bb;
            bb.x = half ? q.y : q.x;   // K = k0 + half
            bb.y = half ? q.w : q.z;   // K = k0 + half + 2
            b[j] = bb;
        }

#pragma unroll
        for (int i = 0; i < 4; ++i)
#pragma unroll
            for (int j = 0; j < 4; ++j)
                acc[i][j] = __builtin_amdgcn_wmma_f32_16x16x4_f32(
                    /*neg_a=*/false, a[i], /*neg_b=*/false, b[j],
                    /*c_mod=*/(short)0, acc[i][j],
                    /*reuse_a=*/false, /*reuse_b=*/false);
    }

    // Epilogue: out = exp(-max(xsq + rsq - 2*cross, 0))
#pragma unroll
    for (int i = 0; i < 4; ++i) {
        float xs[8];
#pragma unroll
        for (int v = 0; v < 8; ++v)
            xs[v] = xsq[rowBase + i * 16 + v + 8 * half];
#pragma unroll
        for (int j = 0; j < 4; ++j) {
            const float rc = rsq[colBase + j * 16 + l];
#pragma unroll
            for (int v = 0; v < 8; ++v) {
                const int row = rowBase + i * 16 + v + 8 * half;
                float d = xs[v] + rc - 2.0f * acc[i][j][v];
                d = fmaxf(d, 0.0f);
                out[(size_t)row * M_COLS + colBase + j * 16 + l] = __expf(-d);
            }
        }
    }
}

extern "C" void kernel_launch(void* const* d_in, const int* in_sizes, int n_in,
                              void* d_out, int out_size, void* d_ws, size_t ws_size,
                              hipStream_t stream) {
    const float* x = (const float*)d_in[0];            // [32768, 64]
    const float* r = (const float*)d_in[1];            // [2048, 64]
    float* out = (float*)d_out;                        // [32768, 2048]

    float* xsq = (float*)d_ws;                         // 32768 floats
    float* rsq = xsq + N_ROWS;                         // 2048 floats

    // Pass 1: row norms
    rbf_sumsq_kernel<<<(N_ROWS + 255) / 256, 256, 0, stream>>>(x, xsq, N_ROWS);
    rbf_sumsq_kernel<<<(M_COLS + 255) / 256, 256, 0, stream>>>(r, rsq, M_COLS);

    // Pass 2: 512*32 = 16384 wave-blocks, 8 waves per workgroup -> 2048 blocks.
    const int totalWaves = ROW_BLKS * COL_BLKS;
    rbf_wmma_kernel<<<totalWaves / WAVES_PER_BLOCK, WAVES_PER_BLOCK * 32, 0, stream>>>(
        x, r, xsq, rsq, out);
}